// SymmetricContraction_36146444763335
// MI455X (gfx1250) — compile-verified
//
#include <hip/hip_runtime.h>

typedef float v2f __attribute__((ext_vector_type(2)));
typedef float v8f __attribute__((ext_vector_type(8)));

#define NSP   10              // number of species (reference constant)
#define KROW  12              // k-slots per (d0,d1) group: 9×C3, 1×C2, 1×C1, 1 pad
#define KTOT  (81 * KROW)     // 972 k entries total
#define MROWS 16              // padded M (output irreps dim 9 -> 16)

// ---------------------------------------------------------------------------
// Kernel 1: build species-sorted node permutation + 16-node tile tables.
// Trivial serial work (N=2048) on one lane; deterministic.
// ---------------------------------------------------------------------------
__global__ void sc_setup(const int* __restrict__ index, int N,
                         int* __restrict__ meta, int* __restrict__ perm,
                         int* __restrict__ tileSpecies, int* __restrict__ tileSlot) {
  if (threadIdx.x != 0) return;
  int cnt[NSP];
  for (int s = 0; s < NSP; ++s) cnt[s] = 0;
  for (int n = 0; n < N; ++n) {
    int s = index[n];
    if (s >= 0 && s < NSP) cnt[s]++;
  }
  int segBase[NSP];
  int base = 0, tiles = 0;
  for (int s = 0; s < NSP; ++s) {
    segBase[s] = base;
    int nt = (cnt[s] + 15) >> 4;
    for (int t = 0; t < nt; ++t) {
      tileSpecies[tiles] = s;
      tileSlot[tiles]    = base + t * 16;
      ++tiles;
    }
    base += nt * 16;
  }
  meta[0] = tiles;
  for (int i = 0; i < base; ++i) perm[i] = -1;
  int pos[NSP];
  for (int s = 0; s < NSP; ++s) pos[s] = 0;
  for (int n = 0; n < N; ++n) {
    int s = index[n];
    if (s >= 0 && s < NSP) perm[segBase[s] + pos[s]++] = n;
  }
}

// ---------------------------------------------------------------------------
// Kernel 2: one block per (channel c, species s).
// Phase 1: contract U(order,ir) with W(order,ir)[s,:,c] into LDS coefficient
//          matrix Ct[16 rows (=output i, zero-padded)][972 k-slots].
// Phase 2: per wave, GEMM 16x16 node tiles with V_WMMA_F32_16X16X4_F32,
//          building B fragments (monomials of x) on the fly in registers.
// ---------------------------------------------------------------------------
__global__ __launch_bounds__(128) void sc_main(
    const float* __restrict__ X,            // node_feats [N][C][9]
    const int*   __restrict__ meta,
    const int*   __restrict__ perm,
    const int*   __restrict__ tileSpecies,
    const int*   __restrict__ tileSlot,
    float*       __restrict__ out, int C,
    const float* __restrict__ U3_0e, const float* __restrict__ W3_0e,
    const float* __restrict__ U3_1o, const float* __restrict__ W3_1o,
    const float* __restrict__ U3_2e, const float* __restrict__ W3_2e,
    const float* __restrict__ U2_0e, const float* __restrict__ W2_0e,
    const float* __restrict__ U2_1o, const float* __restrict__ W2_1o,
    const float* __restrict__ U2_2e, const float* __restrict__ W2_2e,
    const float* __restrict__ U1_0e, const float* __restrict__ W1_0e,
    const float* __restrict__ U1_1o, const float* __restrict__ W1_1o,
    const float* __restrict__ U1_2e, const float* __restrict__ W1_2e)
{
  __shared__ float Ct[MROWS * KTOT];        // 62.2 KB -> 5 blocks per WGP
  const int c   = blockIdx.x;
  const int s   = blockIdx.y;
  const int tid = threadIdx.x;

  // ---------------- Phase 1: build contracted coefficients in LDS ----------
  for (int e = tid; e < MROWS * KTOT; e += 128) {
    const int m = e / KTOT;                 // output row i (0..15; 9..15 pad)
    const int p = e - m * KTOT;
    float v = 0.0f;
    if (m < 9) {
      const int row  = p / KROW;            // row = d0*9 + d1
      const int slot = p - row * KROW;
      const int d0 = row / 9;
      const int d1 = row - d0 * 9;
      const float *u3, *w3, *u2, *w2, *u1, *w1;
      int mul3, mul2, irdim, ii;
      if (m == 0)     { u3=U3_0e; w3=W3_0e; u2=U2_0e; w2=W2_0e; u1=U1_0e; w1=W1_0e; mul3=10; mul2=3; irdim=1; ii=0;   }
      else if (m < 4) { u3=U3_1o; w3=W3_1o; u2=U2_1o; w2=W2_1o; u1=U1_1o; w1=W1_1o; mul3=11; mul2=2; irdim=3; ii=m-1; }
      else            { u3=U3_2e; w3=W3_2e; u2=U2_2e; w2=W2_2e; u1=U1_2e; w1=W1_2e; mul3=13; mul2=3; irdim=5; ii=m-4; }
      if (slot < 9) {                        // order-3: C3[d0,d1,d2=slot, i]
        const int ub = (row * 9 + slot) * mul3 * irdim + ii;
        for (int k = 0; k < mul3; ++k)
          v += u3[ub + k * irdim] * w3[(s * mul3 + k) * C + c];
      } else if (slot == 9) {               // order-2: C2[d0,d1,i]
        const int ub = row * mul2 * irdim + ii;
        for (int k = 0; k < mul2; ++k)
          v += u2[ub + k * irdim] * w2[(s * mul2 + k) * C + c];
      } else if (slot == 10 && d1 == 0) {   // order-1: C1[d0,i] (mul1 == 1)
        v = u1[d0 * irdim + ii] * w1[s * C + c];
      }                                      // slot 11 and rows 9..15 stay 0
    }
    Ct[e] = v;
  }
  __syncthreads();

  // ---------------- Phase 2: WMMA GEMM over 16-node tiles ------------------
  const int  lane = tid & 31;
  const int  wave = tid >> 5;               // 4 waves per block
  const int  n16  = lane & 15;              // A row (=i) and B column (=node)
  const bool hi   = (lane & 16) != 0;       // k-half select for A/B fragments
  const float* Abase = &Ct[n16 * KTOT + (hi ? 2 : 0)];
  const int numTiles = meta[0];

  int mc = 0;
  for (int t = 0; t < numTiles; ++t) {
    if (tileSpecies[t] != s) continue;      // wave-uniform
    const int my = mc++;
    if ((my & 3) != wave) continue;         // distribute tiles across waves

    const int  slotBase = tileSlot[t];
    const int  node  = perm[slotBase + n16];
    const bool valid = node >= 0;
    const int  nodeC = valid ? node : 0;
    const float* xp = X + (nodeC * C + c) * 9;
    float xv[9];
#pragma unroll
    for (int d = 0; d < 9; ++d) xv[d] = valid ? xp[d] : 0.0f;

    // per-lane monomial factors selected once per tile (k-half dependent)
    const float e0 = hi ? xv[2] : xv[0];
    const float e1 = hi ? xv[3] : xv[1];
    const float e2 = hi ? xv[6] : xv[4];
    const float e3 = hi ? xv[7] : xv[5];
    const float e4 = hi ? 0.0f  : xv[8];
    const float hm = hi ? 0.0f  : 1.0f;     // slot9 (order-2, phi = x_d0*x_d1)

    v8f acc0 = {0.f,0.f,0.f,0.f,0.f,0.f,0.f,0.f};
    v8f acc1 = acc0, acc2 = acc0;

#pragma unroll 1
    for (int d1i = 0; d1i < 9; ++d1i) {
      float xd1 = xv[0];
#pragma unroll
      for (int q = 1; q < 9; ++q) if (d1i == q) xd1 = xv[q];
      const int rb = d1i * KROW;
#pragma unroll
      for (int d0 = 0; d0 < 9; ++d0) {
        const float xx = xv[d0] * xd1;      // x_d0 * x_d1
        const float* ap = Abase + d0 * (9 * KROW) + rb;
        v2f af0 = *(const v2f*)(ap);        // k-slots {0,1}/{2,3}
        v2f af1 = *(const v2f*)(ap + 4);    // k-slots {4,5}/{6,7}
        v2f af2 = *(const v2f*)(ap + 8);    // k-slots {8,9}/{10,11}
        v2f b0, b1, b2;
        b0.x = xx * e0;  b0.y = xx * e1;    // phi = x_d0 x_d1 x_d2
        b1.x = xx * e2;  b1.y = xx * e3;
        const float c1h = (hi && d1i == 0) ? xv[d0] : 0.0f;  // slot10: phi=x_d0
        b2.x = xx * e4 + c1h;               // lo: x_d0 x_d1 x_8 ; hi: order-1
        b2.y = xx * hm;                     // lo: order-2 phi   ; hi: pad 0
        acc0 = __builtin_amdgcn_wmma_f32_16x16x4_f32(false, af0, false, b0,
                                                     (short)0, acc0, false, false);
        acc1 = __builtin_amdgcn_wmma_f32_16x16x4_f32(false, af1, false, b1,
                                                     (short)0, acc1, false, false);
        acc2 = __builtin_amdgcn_wmma_f32_16x16x4_f32(false, af2, false, b2,
                                                     (short)0, acc2, false, false);
      }
    }
    v8f acc = acc0 + acc1 + acc2;

    // D layout: lanes 0-15 hold rows i=r, lanes 16-31 hold rows i=r+8
    if (valid) {
      float* op = out + (node * C + c) * 9;
      if (!hi) {
#pragma unroll
        for (int r = 0; r < 8; ++r) op[r] = acc[r];
      } else {
        op[8] = acc[0];
      }
    }
  }
}

// ---------------------------------------------------------------------------
extern "C" void kernel_launch(void* const* d_in, const int* in_sizes, int n_in,
                              void* d_out, int out_size, void* d_ws, size_t ws_size,
                              hipStream_t stream) {
  (void)n_in; (void)out_size; (void)ws_size;
  const float* X     = (const float*)d_in[0];
  const int*   index = (const int*)d_in[1];
  const int N = in_sizes[1];
  const int C = in_sizes[0] / (N * 9);

  // workspace layout (ints): meta[16] | perm[N+16*NSP] | tileSpecies[256] | tileSlot[256]
  int* meta        = (int*)d_ws;
  int* perm        = meta + 16;
  int  permCap     = (N + 16 * NSP + 15) & ~15;
  int* tileSpecies = perm + permCap;
  int* tileSlot    = tileSpecies + 256;

  sc_setup<<<1, 32, 0, stream>>>(index, N, meta, perm, tileSpecies, tileSlot);

  sc_main<<<dim3(C, NSP), 128, 0, stream>>>(
      X, meta, perm, tileSpecies, tileSlot, (float*)d_out, C,
      (const float*)d_in[2],  (const float*)d_in[3],   // U3_0e, W3_0e
      (const float*)d_in[4],  (const float*)d_in[5],   // U3_1o, W3_1o
      (const float*)d_in[6],  (const float*)d_in[7],   // U3_2e, W3_2e
      (const float*)d_in[8],  (const float*)d_in[9],   // U2_0e, W2_0e
      (const float*)d_in[10], (const float*)d_in[11],  // U2_1o, W2_1o
      (const float*)d_in[12], (const float*)d_in[13],  // U2_2e, W2_2e
      (const float*)d_in[14], (const float*)d_in[15],  // U1_0e, W1_0e
      (const float*)d_in[16], (const float*)d_in[17],  // U1_1o, W1_1o
      (const float*)d_in[18], (const float*)d_in[19]); // U1_2e, W1_2e
}